// SelectedUnitsHead_47236050321700
// MI455X (gfx1250) — compile-verified
//
#include <hip/hip_runtime.h>
#include <hip/hip_bf16.h>
#include <math.h>
#include <stdint.h>

typedef __attribute__((ext_vector_type(2))) float v2f;
typedef __attribute__((ext_vector_type(8))) float v8f;

#define KC 32
#define APAD 4   // LDS row padding (floats); keeps 16B alignment for async B128

// Async global->LDS copy of 16 bytes (CDNA5 GLOBAL_LOAD_ASYNC_TO_LDS_B128,
// tracked by ASYNCcnt; no VGPR round-trip, no s_wait_loadcnt serialization).
__device__ __forceinline__ void async_copy_b128(void* lds_ptr, const void* gptr) {
    unsigned lds = (unsigned)(uintptr_t)lds_ptr;           // low 32b = LDS offset
    unsigned long long ga = (unsigned long long)(uintptr_t)gptr;
    asm volatile("global_load_async_to_lds_b128 %0, %1, off"
                 :: "v"(lds), "v"(ga) : "memory");
}
__device__ __forceinline__ void wait_async_all() {
    asm volatile("s_wait_asynccnt 0x0" ::: "memory");
}

// ---------------------------------------------------------------------------
// f32 WMMA GEMM:  O = act(A[M,K] @ W[K,N] + bias) [+ resid[row%mod]]
// 8 waves arranged WR x WC, each wave owns a 32x32 output (2x2 grid of 16x16
// V_WMMA_F32_16X16X4_F32 accumulators) so each LDS fragment feeds 2 wmma.
// Tiles are staged with async global->LDS B128 transfers; fragments are
// software-pipelined one K-step ahead so ds_load latency overlaps wmma issue.
// Requires: M % (WR*32) == 0, N % (WC*32) == 0, K % 32 == 0, lda/ldw % 4 == 0.
// row_pad_every!=0 remaps output row r -> r + r/row_pad_every (key padding).
// ---------------------------------------------------------------------------
__device__ __forceinline__ void store_tile16(
    v8f acc, float bn, int m0, int n, int hl,
    float* __restrict__ O, int ldo,
    const float* __restrict__ resid, int resid_mod, int ldr,
    int do_relu, int row_pad_every)
{
#pragma unroll
    for (int r = 0; r < 8; ++r) {
        int m   = m0 + r + hl * 8;
        float v = acc[r] + bn;
        if (do_relu) v = fmaxf(v, 0.f);
        if (resid) v += resid[(size_t)(m % resid_mod) * ldr + n];
        int orow = row_pad_every ? (m + m / row_pad_every) : m;
        O[(size_t)orow * ldo + n] = v;
    }
}

template <int WR, int WC>
__global__ __launch_bounds__(256) void gemm_f32_wmma(
    const float* __restrict__ A, int lda,
    const float* __restrict__ W, int ldw,
    const float* __restrict__ bias,
    float* __restrict__ O, int ldo,
    const float* __restrict__ resid, int resid_mod, int ldr,
    int K, int do_relu, int row_pad_every)
{
    constexpr int BMT = WR * 32;
    constexpr int BNT = WC * 32;
    __shared__ float As[BMT][KC + APAD];
    __shared__ float Ws[KC][BNT + APAD];

    const int tid  = threadIdx.x;
    const int wave = tid >> 5;
    const int lane = tid & 31;
    const int hl   = lane >> 4;     // 0: K pair {0,1}, 1: K pair {2,3}
    const int lm   = lane & 15;
    const int m_blk = blockIdx.x * BMT;
    const int n_blk = blockIdx.y * BNT;
    const int wm = (wave % WR) * 32;
    const int wn = (wave / WR) * 32;

    v8f acc00 = {0.f,0.f,0.f,0.f,0.f,0.f,0.f,0.f};
    v8f acc01 = acc00, acc10 = acc00, acc11 = acc00;

    // Fragment loader: A frags for rows wm/wm+16, B frags for cols wn/wn+16.
    // f32 16x16x4 layout: lanes 0-15 hold K pair {kk,kk+1}, lanes 16-31 {kk+2,kk+3}.
    auto loadf = [&](int kk, v2f& A0, v2f& A1, v2f& B0, v2f& B1) {
        int ka = kk + 2 * hl;
        A0.x = As[wm + lm][ka];        A0.y = As[wm + lm][ka + 1];
        A1.x = As[wm + 16 + lm][ka];   A1.y = As[wm + 16 + lm][ka + 1];
        B0.x = Ws[ka][wn + lm];        B0.y = Ws[ka + 1][wn + lm];
        B1.x = Ws[ka][wn + 16 + lm];   B1.y = Ws[ka + 1][wn + 16 + lm];
    };

    for (int k0 = 0; k0 < K; k0 += KC) {
        // Stage A tile (BMT x 32) via async B128: BMT*8 slots, coalesced along K
#pragma unroll
        for (int j = 0; j < (BMT * 8) / 256; ++j) {
            int idx4 = tid + j * 256;
            int m    = idx4 >> 3;
            int kq   = (idx4 & 7) << 2;
            async_copy_b128(&As[m][kq],
                            A + (size_t)(m_blk + m) * lda + k0 + kq);
        }
        // Stage W tile (32 x BNT) via async B128: BNT*8 slots
#pragma unroll
        for (int j = 0; j < (BNT * 8) / 256; ++j) {
            int idx4 = tid + j * 256;
            int k    = idx4 / (BNT / 4);
            int nq   = (idx4 % (BNT / 4)) << 2;
            async_copy_b128(&Ws[k][nq],
                            W + (size_t)(k0 + k) * ldw + n_blk + nq);
        }
        wait_async_all();     // own async transfers landed in LDS
        __syncthreads();      // all waves' tiles visible

        // 8 K-steps of 4, each: 4 fragment pairs -> 4 wmma (2x2 tile reuse),
        // with next step's fragments prefetched before issuing wmma.
        v2f a0, a1, b0, b1, na0, na1, nb0, nb1;
        loadf(0, a0, a1, b0, b1);
#pragma unroll
        for (int kk = 0; kk < KC; kk += 4) {
            if (kk + 4 < KC) loadf(kk + 4, na0, na1, nb0, nb1);
            acc00 = __builtin_amdgcn_wmma_f32_16x16x4_f32(
                false, a0, false, b0, (short)0, acc00, false, false);
            acc01 = __builtin_amdgcn_wmma_f32_16x16x4_f32(
                false, a0, false, b1, (short)0, acc01, false, false);
            acc10 = __builtin_amdgcn_wmma_f32_16x16x4_f32(
                false, a1, false, b0, (short)0, acc10, false, false);
            acc11 = __builtin_amdgcn_wmma_f32_16x16x4_f32(
                false, a1, false, b1, (short)0, acc11, false, false);
            if (kk + 4 < KC) { a0 = na0; a1 = na1; b0 = nb0; b1 = nb1; }
        }
        __syncthreads();
    }

    // Epilogue. C/D layout: VGPR r, lanes 0-15 -> M=r, lanes 16-31 -> M=r+8.
    const int n0 = n_blk + wn + lm;
    const int n1 = n0 + 16;
    const float bn0 = bias ? bias[n0] : 0.f;
    const float bn1 = bias ? bias[n1] : 0.f;
    store_tile16(acc00, bn0, m_blk + wm,      n0, hl, O, ldo, resid, resid_mod, ldr, do_relu, row_pad_every);
    store_tile16(acc01, bn1, m_blk + wm,      n1, hl, O, ldo, resid, resid_mod, ldr, do_relu, row_pad_every);
    store_tile16(acc10, bn0, m_blk + wm + 16, n0, hl, O, ldo, resid, resid_mod, ldr, do_relu, row_pad_every);
    store_tile16(acc11, bn1, m_blk + wm + 16, n1, hl, O, ldo, resid, resid_mod, ldr, do_relu, row_pad_every);
}

// key[b][512][:] = 0 (pad slot); key[b][entity_num[b]][:] = end_embedding
__global__ __launch_bounds__(32) void key_finalize(
    float* __restrict__ key, const int* __restrict__ en,
    const float* __restrict__ end_emb)
{
    int b = blockIdx.x, t = threadIdx.x;
    key[((size_t)b * 513 + 512) * 32 + t] = 0.f;
    key[((size_t)b * 513 + en[b]) * 32 + t] = end_emb[t];
}

// first_sel[b][k] = first step s with selected_units[b][s]==k (else 2^30)
__global__ __launch_bounds__(256) void first_sel_kernel(
    const int* __restrict__ su, int* __restrict__ fsel)
{
    int b = blockIdx.x;
    for (int k = threadIdx.x; k < 513; k += 256) fsel[b * 513 + k] = 1 << 30;
    __syncthreads();
    if (threadIdx.x < 64)
        atomicMin(&fsel[b * 513 + su[b * 64 + threadIdx.x]], (int)threadIdx.x);
}

// Incremental one_hot/emb evolution (teacher forced; independent of LSTM).
// embdiv[s][b][0:32] = (sum of distinct pre-end selected key rows) / denom
__global__ __launch_bounds__(32) void emb_seq_kernel(
    const float* __restrict__ key, const int* __restrict__ su,
    const int* __restrict__ en, const int* __restrict__ sun,
    float* __restrict__ embdiv)
{
    int b = blockIdx.x, t = threadIdx.x;
    __shared__ unsigned char bm[513];
    __shared__ int s_new;
    for (int i = t; i < 513; i += 32) bm[i] = 0;
    float sum = 0.f;
    int cnt = 0;
    bool endf = false;
    const int enb = en[b];
    const bool sun_nz = (sun[b] != 0);
    for (int s = 0; s < 64; ++s) {
        int u = su[b * 64 + s];
        endf = endf || (u == enb);     // end_flag updated BEFORE one_hot write
        __syncthreads();
        if (t == 0) {
            int nw = (!endf && !bm[u]) ? 1 : 0;
            if (nw) bm[u] = 1;
            s_new = nw;
        }
        __syncthreads();
        if (s_new) { sum += key[((size_t)b * 513 + u) * 32 + t]; cnt++; }
        float denom = (sun_nz && cnt > 0) ? (float)cnt : 1.0f;
        embdiv[((size_t)s * 512 + b) * 32 + t] = sum / denom;
    }
}

__global__ __launch_bounds__(256) void copy_ae0(
    const float* __restrict__ ar, float* __restrict__ ae)
{
    int i = blockIdx.x * 256 + threadIdx.x;   // 512*1024 total
    ae[i] = ar[i];
}

__device__ __forceinline__ float block_sum_128(float v, float* red) {
    int j = threadIdx.x;
    red[j] = v;
    __syncthreads();
#pragma unroll
    for (int off = 64; off > 0; off >>= 1) {
        if (j < off) red[j] += red[j + off];
        __syncthreads();
    }
    float r = red[0];
    __syncthreads();
    return r;
}

// Sequential LayerNorm-LSTM cell over 64 steps, one block per batch element.
__global__ __launch_bounds__(128) void lstm_seq(
    const float* __restrict__ x_all,
    const float* __restrict__ w_ih, const float* __restrict__ w_hh,
    const float* __restrict__ ln_ig, const float* __restrict__ ln_ib,
    const float* __restrict__ ln_hg, const float* __restrict__ ln_hb,
    const float* __restrict__ ln_cg, const float* __restrict__ ln_cb,
    float* __restrict__ queries)
{
    const int b = blockIdx.x;
    const int j = threadIdx.x;           // gate index 0..127
    __shared__ float sWih[32 * 128];
    __shared__ float sWhh[32 * 128];
    __shared__ float sx[32], sh[32], sc[32];
    __shared__ float red[128];
    __shared__ float gsh[128];
    __shared__ float cand[32];
    for (int i = j; i < 32 * 128; i += 128) { sWih[i] = w_ih[i]; sWhh[i] = w_hh[i]; }
    if (j < 32) { sh[j] = 0.f; sc[j] = 0.f; }
    const float gig = ln_ig[j], gib = ln_ib[j];
    const float ghg = ln_hg[j], ghb = ln_hb[j];
    const float gcg = (j < 32) ? ln_cg[j] : 0.f;
    const float gcb = (j < 32) ? ln_cb[j] : 0.f;
    __syncthreads();

    for (int s = 0; s < 64; ++s) {
        if (j < 32) sx[j] = x_all[((size_t)s * 512 + b) * 32 + j];
        __syncthreads();
        float gx = 0.f, gh = 0.f;
#pragma unroll
        for (int d = 0; d < 32; ++d) {
            gx = fmaf(sx[d], sWih[d * 128 + j], gx);
            gh = fmaf(sh[d], sWhh[d * 128 + j], gh);
        }
        float mx = block_sum_128(gx, red) * (1.f / 128.f);
        float vx = block_sum_128(gx * gx, red) * (1.f / 128.f) - mx * mx;
        float nx = (gx - mx) * rsqrtf(vx + 1e-5f) * gig + gib;
        float mh = block_sum_128(gh, red) * (1.f / 128.f);
        float vh = block_sum_128(gh * gh, red) * (1.f / 128.f) - mh * mh;
        float nh = (gh - mh) * rsqrtf(vh + 1e-5f) * ghg + ghb;
        gsh[j] = nx + nh;
        __syncthreads();
        if (j < 32) {
            float ig = gsh[j], fg = gsh[32 + j], cg = gsh[64 + j];
            cand[j] = (1.f / (1.f + expf(-fg))) * sc[j]
                    + (1.f / (1.f + expf(-ig))) * tanhf(cg);
        }
        __syncthreads();
        if (j < 32) {
            float t1 = 0.f, t2 = 0.f;
#pragma unroll
            for (int d = 0; d < 32; ++d) { float x = cand[d]; t1 += x; t2 += x * x; }
            float m = t1 * (1.f / 32.f);
            float v = t2 * (1.f / 32.f) - m * m;
            float cn = (cand[j] - m) * rsqrtf(v + 1e-5f) * gcg + gcb;
            float hn = (1.f / (1.f + expf(-gsh[96 + j]))) * tanhf(cn);
            sc[j] = cn;
            sh[j] = hn;
            queries[((size_t)s * 512 + b) * 32 + j] = hn;
        }
        __syncthreads();
    }
}

// logits[b][s][n] = valid(b,s,n) ? dot(queries[s][b], key[b][n]) : -1e9
// Mask reconstructed analytically from first_sel / entity_num.
__global__ __launch_bounds__(256) void logits_kernel(
    const float* __restrict__ queries, const float* __restrict__ key,
    const int* __restrict__ fsel, const int* __restrict__ en,
    float* __restrict__ out)
{
    const int b = blockIdx.x;
    __shared__ float q[64 * 32];       // all 64 step queries for this batch
    for (int i = threadIdx.x; i < 64 * 32; i += 256)
        q[i] = queries[((size_t)(i >> 5) * 512 + b) * 32 + (i & 31)];
    __syncthreads();
    const int enb = en[b];
    for (int n = threadIdx.x; n < 513; n += 256) {
        float krow[32];
        const float* kp = key + ((size_t)b * 513 + n) * 32;
#pragma unroll
        for (int d = 0; d < 32; ++d) krow[d] = kp[d];
        const int fs = fsel[b * 513 + n];
        for (int s = 0; s < 64; ++s) {
            bool valid;
            if (n < enb)       valid = (fs >= s);
            else if (n == enb) valid = (s >= 1) && (fs >= s);
            else               valid = false;
            float dot = 0.f;
#pragma unroll
            for (int d = 0; d < 32; ++d) dot = fmaf(q[s * 32 + d], krow[d], dot);
            out[((size_t)b * 64 + s) * 513 + n] = valid ? dot : -1e9f;
        }
    }
}

extern "C" void kernel_launch(void* const* d_in, const int* in_sizes, int n_in,
                              void* d_out, int out_size, void* d_ws, size_t ws_size,
                              hipStream_t stream) {
    const float* ent  = (const float*)d_in[0];   // [512,512,256]
    const float* arte = (const float*)d_in[1];   // [512,1024]
    const int*   en   = (const int*)d_in[2];     // [512]
    const int*   su   = (const int*)d_in[3];     // [512,64]
    const int*   sun  = (const int*)d_in[4];     // [512]
    const float* kfw  = (const float*)d_in[5];   // [256,32]
    const float* kfb  = (const float*)d_in[6];
    const float* q1w  = (const float*)d_in[7];   // [1024,256]
    const float* q1b  = (const float*)d_in[8];
    const float* q2w  = (const float*)d_in[9];   // [256,32]
    const float* q2b  = (const float*)d_in[10];
    const float* e1w  = (const float*)d_in[11];  // [32,256]
    const float* e1b  = (const float*)d_in[12];
    const float* e2w  = (const float*)d_in[13];  // [256,1024]
    const float* e2b  = (const float*)d_in[14];
    const float* eemb = (const float*)d_in[15];  // [32]
    const float* wih  = (const float*)d_in[16];
    const float* whh  = (const float*)d_in[17];
    const float* lnig = (const float*)d_in[18];
    const float* lnib = (const float*)d_in[19];
    const float* lnhg = (const float*)d_in[20];
    const float* lnhb = (const float*)d_in[21];
    const float* lncg = (const float*)d_in[22];
    const float* lncb = (const float*)d_in[23];
    float* out = (float*)d_out;

    // Workspace carve-up (f32 unless noted)
    float* key    = (float*)d_ws;                       // 512*513*32
    float* embdiv = key    + (size_t)512 * 513 * 32;    // 64*512*32
    float* tbuf   = embdiv + (size_t)64 * 512 * 32;     // 64*512*256 (shared)
    float* ae     = tbuf   + (size_t)64 * 512 * 256;    // 64*512*1024
    float* xall   = ae     + (size_t)64 * 512 * 1024;   // 64*512*32
    float* qrs    = xall   + (size_t)64 * 512 * 32;     // 64*512*32
    int*   fsel   = (int*)(qrs + (size_t)64 * 512 * 32);// 512*513 ints

    // 1) key = ent @ key_fc_w + b, written into padded [512,513,32] layout
    gemm_f32_wmma<8, 1><<<dim3(262144 / 256, 1), 256, 0, stream>>>(
        ent, 256, kfw, 32, kfb, key, 32, nullptr, 1, 1, 256, 0, 512);
    key_finalize<<<512, 32, 0, stream>>>(key, en, eemb);

    // 2) mask precompute + incremental emb sequence
    first_sel_kernel<<<512, 256, 0, stream>>>(su, fsel);
    emb_seq_kernel<<<512, 32, 0, stream>>>(key, su, en, sun, embdiv);

    // 3) e-path batched over steps 0..62 -> ae for steps 1..63 (M=32256)
    gemm_f32_wmma<4, 2><<<dim3(32256 / 128, 256 / 64), 256, 0, stream>>>(
        embdiv, 32, e1w, 256, e1b, tbuf, 256, nullptr, 1, 1, 32, 1, 0);
    gemm_f32_wmma<4, 2><<<dim3(32256 / 128, 1024 / 64), 256, 0, stream>>>(
        tbuf, 256, e2w, 1024, e2b, ae + (size_t)512 * 1024, 1024,
        arte, 512, 1024, 256, 0, 0);
    copy_ae0<<<2048, 256, 0, stream>>>(arte, ae);       // ae step 0 = input

    // 4) q-path batched over all 64 steps (M=32768)
    gemm_f32_wmma<4, 2><<<dim3(32768 / 128, 256 / 64), 256, 0, stream>>>(
        ae, 1024, q1w, 256, q1b, tbuf, 256, nullptr, 1, 1, 1024, 1, 0);
    gemm_f32_wmma<8, 1><<<dim3(32768 / 256, 1), 256, 0, stream>>>(
        tbuf, 256, q2w, 32, q2b, xall, 32, nullptr, 1, 1, 256, 0, 0);

    // 5) sequential LN-LSTM (only truly serial work; tiny per step)
    lstm_seq<<<512, 128, 0, stream>>>(xall, wih, whh, lnig, lnib,
                                      lnhg, lnhb, lncg, lncb, qrs);

    // 6) masked logits, key read once per batch
    logits_kernel<<<512, 256, 0, stream>>>(qrs, key, fsel, en, out);
}